// MSEOHEMLoss_79061757985022
// MI455X (gfx1250) — compile-verified
//
#include <hip/hip_runtime.h>

// Clang ext vector types (HIP's float4 is a struct; these lower cleanly to
// global_load_b128 / b64 and accept __builtin_nontemporal_load).
typedef __attribute__((ext_vector_type(4))) float f4;
typedef __attribute__((ext_vector_type(2))) float f2;
typedef __attribute__((ext_vector_type(2))) float v2f;
typedef __attribute__((ext_vector_type(8))) float v8f;

// Exact fp32 reduction of 32 per-lane partials using V_WMMA_F32_16X16X4_F32.
// A (16x4): lane L<16 holds (M=L, K=0..1), lane L>=16 holds (M=L-16, K=2..3).
// We place the partial in the lane's first A slot and 0 in the second.
// B (4x16) = all ones  =>  D[m][n] = sum_k A[m][k]  (same for every n).
// D layout: lanes 0..15 hold M=0..7 in VGPR0..7, lanes 16..31 hold M=8..15.
// => (sum of 8 D elems at lane 0) + (same at lane 16) == sum of all 32 partials.
__device__ __forceinline__ float wave_reduce_wmma(float acc) {
    v2f a; a[0] = acc;  a[1] = 0.0f;
    v2f b; b[0] = 1.0f; b[1] = 1.0f;
    v8f c = {};
    // 8 args: (neg_a, A, neg_b, B, c_mod, C, reuse_a, reuse_b)
    v8f d = __builtin_amdgcn_wmma_f32_16x16x4_f32(
        /*neg_a=*/false, a, /*neg_b=*/false, b,
        /*c_mod=*/(short)0, c, /*reuse_a=*/false, /*reuse_b=*/false);
    float s = d[0] + d[1] + d[2] + d[3] + d[4] + d[5] + d[6] + d[7];
    return __shfl(s, 0, 32) + __shfl(s, 16, 32);
}

// Pass 1: streaming squared-error partial sums, one fp32 partial per block.
// pred is [B,H,W,2] interleaved (char,aff); cg/ag are [B,H,W].
__global__ void mse_partial_kernel(const float* __restrict__ pred,
                                   const float* __restrict__ cg,
                                   const float* __restrict__ ag,
                                   float* __restrict__ partials,
                                   long long nPix) {
    const long long nPair  = nPix >> 1;          // process 2 pixels / iteration
    const long long tid    = (long long)blockIdx.x * blockDim.x + threadIdx.x;
    const long long stride = (long long)gridDim.x * blockDim.x;

    const f4* __restrict__ p4 = (const f4*)pred; // 2 pixels: (c0,a0,c1,a1)
    const f2* __restrict__ c2 = (const f2*)cg;
    const f2* __restrict__ a2 = (const f2*)ag;

    float acc = 0.0f;
    for (long long i = tid; i < nPair; i += stride) {
        f4 p = __builtin_nontemporal_load(&p4[i]);   // 16B, TH=NT (no reuse, > L2)
        f2 c = __builtin_nontemporal_load(&c2[i]);   // 8B
        f2 a = __builtin_nontemporal_load(&a2[i]);   // 8B
        float d0 = p[0] - c[0];
        float d1 = p[1] - a[0];
        float d2 = p[2] - c[1];
        float d3 = p[3] - a[1];
        acc = fmaf(d0, d0, acc);
        acc = fmaf(d1, d1, acc);
        acc = fmaf(d2, d2, acc);
        acc = fmaf(d3, d3, acc);
    }
    // odd-pixel tail (not hit for 16*1024*1024, but keep it general)
    if ((nPix & 1LL) && tid == 0) {
        long long last = nPix - 1;
        float d0 = pred[2 * last]     - cg[last];
        float d1 = pred[2 * last + 1] - ag[last];
        acc = fmaf(d0, d0, acc);
        acc = fmaf(d1, d1, acc);
    }

    // Uniform control flow here: EXEC is all-ones for the WMMA.
    float w = wave_reduce_wmma(acc);

    __shared__ float ws[8];                       // 256 threads = 8 wave32s
    const int lane = threadIdx.x & 31;
    const int wid  = threadIdx.x >> 5;
    if (lane == 0) ws[wid] = w;
    __syncthreads();
    if (threadIdx.x == 0) {
        float t = 0.0f;
        const int nw = (blockDim.x + 31) >> 5;
        for (int i = 0; i < nw; ++i) t += ws[i];
        partials[blockIdx.x] = t;
    }
}

// Pass 2: single block, deterministic fixed-order reduction + scale by 1/N.
__global__ void mse_final_kernel(const float* __restrict__ partials, int n,
                                 float* __restrict__ out, float invN) {
    float acc = 0.0f;
    for (int i = threadIdx.x; i < n; i += blockDim.x) acc += partials[i];

    float w = wave_reduce_wmma(acc);

    __shared__ float ws[32];
    const int lane = threadIdx.x & 31;
    const int wid  = threadIdx.x >> 5;
    if (lane == 0) ws[wid] = w;
    __syncthreads();
    if (threadIdx.x == 0) {
        float t = 0.0f;
        const int nw = blockDim.x >> 5;
        for (int i = 0; i < nw; ++i) t += ws[i];
        out[0] = t * invN;
    }
}

extern "C" void kernel_launch(void* const* d_in, const int* in_sizes, int n_in,
                              void* d_out, int out_size, void* d_ws, size_t ws_size,
                              hipStream_t stream) {
    const float* pred = (const float*)d_in[0];   // [B,H,W,2] fp32
    const float* cg   = (const float*)d_in[1];   // [B,H,W]   fp32
    const float* ag   = (const float*)d_in[2];   // [B,H,W]   fp32
    float* out = (float*)d_out;

    const long long nPix = (long long)in_sizes[1];   // B*H*W

    // 2048 blocks x 256 threads = 16K wave32s: enough in-flight b128/b64
    // loads to saturate 23.3 TB/s. Clamp partial count to workspace size.
    int blocks = 2048;
    const int maxPartials = (int)(ws_size / sizeof(float));
    if (maxPartials > 0 && blocks > maxPartials) blocks = maxPartials;
    if (blocks < 1) blocks = 1;

    float* partials = (float*)d_ws;

    mse_partial_kernel<<<blocks, 256, 0, stream>>>(pred, cg, ag, partials, nPix);
    mse_final_kernel<<<1, 1024, 0, stream>>>(partials, blocks, out,
                                             1.0f / (float)nPix);
}